// CGRCNet_25357486915682
// MI455X (gfx1250) — compile-verified
//
#include <hip/hip_runtime.h>

typedef float v2f __attribute__((ext_vector_type(2)));
typedef float v8f __attribute__((ext_vector_type(8)));

#define U_DIM 2048
#define D_DIM 64
#define CD_DIM 300
#define C_DIM 1024
#define H_DIM 128

// ---------------------------------------------------------------------------
// Generic f32 WMMA GEMM: out[m,n] = sum_k A[row(m), k0+k] * B[n*ldb + koff + k]
//                                   + (bias ? bias[n] : 0)
// row(m) = gather ? gather[m] : m.  One wave per 16x16 output tile, using
// V_WMMA_F32_16X16X4_F32 (f32 in / f32 accumulate — matches reference dtype).
//
// Per CDNA5 ISA layouts (CDNA5 ISA):
//   A 16x4:  lane = M (lane&15), components = K pair at offset 2*(lane>>4)
//   B 4x16:  lane = N (lane&15), components = K pair at offset 2*(lane>>4)
//   C/D:     vgpr i -> M = i + 8*(lane>>4), N = lane&15
// ---------------------------------------------------------------------------
__global__ __launch_bounds__(128)
void wmma_gemm_f32_kernel(const float* __restrict__ A, int lda,
                          const int* __restrict__ gather,
                          const float* __restrict__ B, int ldb, int koff,
                          const float* __restrict__ bias,
                          float* __restrict__ out, int ldo,
                          int nTiles, int totalTiles, int K)
{
    int wave = threadIdx.x >> 5;
    int tile = blockIdx.x * (blockDim.x >> 5) + wave;
    if (tile >= totalTiles) return;          // wave-uniform: EXEC stays all-1s

    int mT   = tile / nTiles;
    int nT   = tile % nTiles;
    int lane = threadIdx.x & 31;
    int half = lane >> 4;                    // selects K pair {0,1} vs {2,3}
    int lm   = lane & 15;

    int m    = mT * 16 + lm;
    int arow = gather ? gather[m] : m;
    const float* ap = A + (long)arow * lda + half * 2;

    int n = nT * 16 + lm;
    const float* bp = B + (long)n * ldb + koff + half * 2;

    v8f acc = {0.f, 0.f, 0.f, 0.f, 0.f, 0.f, 0.f, 0.f};
    for (int k = 0; k < K; k += 4) {
        v2f a = *(const v2f*)(ap + k);       // 8B loads, 8B-aligned for all ld*
        v2f b = *(const v2f*)(bp + k);
        acc = __builtin_amdgcn_wmma_f32_16x16x4_f32(
            /*neg_a=*/false, a, /*neg_b=*/false, b,
            /*c_mod=*/(short)0, acc, /*reuse_a=*/false, /*reuse_b=*/false);
    }

    float bv = bias ? bias[n] : 0.f;
#pragma unroll
    for (int i = 0; i < 8; ++i) {
        int row = mT * 16 + half * 8 + i;
        out[(long)row * ldo + n] = acc[i] + bv;
    }
}

// ---------------------------------------------------------------------------
// Fused logits: out[u,c] = sum_h relu(HU[u,h] + HX[c,h]) * W2[h] + b2
// (b1 already folded into HU, bi into HX's producer). 64u x 32c tile per
// 256-thread block; LDS stride 129 -> bank-conflict-free reads.
// ---------------------------------------------------------------------------
#define TU 64
#define TC 32
#define LSTR 129

__global__ __launch_bounds__(256)
void fused_logits_kernel(const float* __restrict__ HU,   // [U_DIM][H_DIM]
                         const float* __restrict__ HX,   // [C_DIM][H_DIM]
                         const float* __restrict__ W2,   // [H_DIM]
                         const float* __restrict__ b2,   // [1]
                         float* __restrict__ out)        // [U_DIM][C_DIM]
{
    __shared__ float sU[TU * LSTR];
    __shared__ float sX[TC * LSTR];
    __shared__ float sW[H_DIM];

    int t     = threadIdx.x;
    int uBase = blockIdx.y * TU;
    int cBase = blockIdx.x * TC;

    // Stage HU tile: 64*128 floats = 8 float4 per thread, coalesced.
#pragma unroll
    for (int v = 0; v < (TU * H_DIM) / (256 * 4); ++v) {
        int lin = (v * 256 + t) * 4;
        int r = lin >> 7, hc = lin & 127;
        float4 d = *(const float4*)(HU + (uBase + r) * H_DIM + hc);
        float* p = &sU[r * LSTR + hc];
        p[0] = d.x; p[1] = d.y; p[2] = d.z; p[3] = d.w;
    }
    // Stage HX tile: 32*128 floats = 4 float4 per thread.
#pragma unroll
    for (int v = 0; v < (TC * H_DIM) / (256 * 4); ++v) {
        int lin = (v * 256 + t) * 4;
        int r = lin >> 7, hc = lin & 127;
        float4 d = *(const float4*)(HX + (cBase + r) * H_DIM + hc);
        float* p = &sX[r * LSTR + hc];
        p[0] = d.x; p[1] = d.y; p[2] = d.z; p[3] = d.w;
    }
    if (t < H_DIM) sW[t] = W2[t];
    __syncthreads();

    int tx = t & 15;          // c lane:  c = cBase + tx + 16*j
    int ty = t >> 4;          // u lane:  u = uBase + ty + 16*i

    float acc[4][2] = {{0.f, 0.f}, {0.f, 0.f}, {0.f, 0.f}, {0.f, 0.f}};

    for (int h = 0; h < H_DIM; ++h) {
        float w = sW[h];
        float uv[4], cv[2];
#pragma unroll
        for (int i = 0; i < 4; ++i) uv[i] = sU[(ty + 16 * i) * LSTR + h];
#pragma unroll
        for (int j = 0; j < 2; ++j) cv[j] = sX[(tx + 16 * j) * LSTR + h];
#pragma unroll
        for (int i = 0; i < 4; ++i)
#pragma unroll
            for (int j = 0; j < 2; ++j)
                acc[i][j] += fmaxf(uv[i] + cv[j], 0.f) * w;
    }

    float b2v = b2[0];
#pragma unroll
    for (int i = 0; i < 4; ++i)
#pragma unroll
        for (int j = 0; j < 2; ++j)
            out[(long)(uBase + ty + 16 * i) * C_DIM + (cBase + tx + 16 * j)] =
                acc[i][j] + b2v;
}

// ---------------------------------------------------------------------------
extern "C" void kernel_launch(void* const* d_in, const int* in_sizes, int n_in,
                              void* d_out, int out_size, void* d_ws, size_t ws_size,
                              hipStream_t stream)
{
    const float* h_u_bar      = (const float*)d_in[0];   // (2048, 64)
    const float* item_content = (const float*)d_in[1];   // (50000, 300)
    const int*   cold_ids     = (const int*)  d_in[2];   // (1024,)
    const float* Wi           = (const float*)d_in[3];   // (64, 300)
    const float* bi           = (const float*)d_in[4];   // (64,)
    const float* W1           = (const float*)d_in[5];   // (128, 128)
    const float* b1           = (const float*)d_in[6];   // (128,)
    const float* W2           = (const float*)d_in[7];   // (1, 128)
    const float* b2           = (const float*)d_in[8];   // (1,)
    float*       out          = (float*)d_out;           // (2048, 1024)

    float* XC = (float*)d_ws;            // 1024 x 64
    float* HU = XC + C_DIM * D_DIM;      // 2048 x 128  (b1 folded in)
    float* HX = HU + U_DIM * H_DIM;      // 1024 x 128

    // XC = item_content[cold_ids] @ Wi^T + bi   (gather BEFORE gemm: 50x less work)
    // tiles: (1024/16) * (64/16) = 64*4 = 256
    wmma_gemm_f32_kernel<<<64, 128, 0, stream>>>(
        item_content, CD_DIM, cold_ids, Wi, CD_DIM, 0, bi,
        XC, D_DIM, /*nTiles=*/4, /*totalTiles=*/256, /*K=*/CD_DIM);

    // HU = h_u_bar @ W1[:, :64]^T + b1          tiles: 128*8 = 1024
    wmma_gemm_f32_kernel<<<256, 128, 0, stream>>>(
        h_u_bar, D_DIM, nullptr, W1, 2 * D_DIM, 0, b1,
        HU, H_DIM, /*nTiles=*/8, /*totalTiles=*/1024, /*K=*/D_DIM);

    // HX = XC @ W1[:, 64:]^T                    tiles: 64*8 = 512
    wmma_gemm_f32_kernel<<<128, 128, 0, stream>>>(
        XC, D_DIM, nullptr, W1, 2 * D_DIM, D_DIM, nullptr,
        HX, H_DIM, /*nTiles=*/8, /*totalTiles=*/512, /*K=*/D_DIM);

    // logits = relu(HU[:,None,:] + HX[None,:,:]) . W2 + b2   (fully fused)
    dim3 grid(C_DIM / TC, U_DIM / TU);   // (32, 32)
    fused_logits_kernel<<<grid, 256, 0, stream>>>(HU, HX, W2, b2, out);
}